// SparsTriangularUpdate_82128364634682
// MI455X (gfx1250) — compile-verified
//
#include <hip/hip_runtime.h>
#include <hip/hip_bf16.h>

// ---------------------------------------------------------------------------
// SparsTriangularUpdate for MI455X (gfx1250), fp32 WMMA path, v3.
//
// B=2, N=65536, D=C=128, 4096 active rows (stride 16), DEG=16 ring.
// Analytic collapse of the pair structure:
//   p[b,s,:]       = a[b,16s,:] * b[b,16s,:]              (s in [0,4096))
//   k[b,16t+d-1,:] = sum_{u=d+1..16} p[b,(t+u)%4096,:]    (d in [1,16])
// so k is dense over all N rows and the scatter is a ring suffix-sum.
//
// v3: weights staged row-major with PAD=132 stride. B fragment = contiguous
// aligned b64 at (n0+l16)*PAD + kSub + kg*16B: one base add per 16x16 tile,
// all 32 k-group loads fit the ds_load_2addr_b64 immediate-offset window,
// and the 32-lane fragment read covers all 64 LDS banks exactly once.
// ---------------------------------------------------------------------------

#define NN      65536
#define DD      128
#define NACT    4096
#define PAD     132                  // LDS row stride (floats); mod 64 banks = 4
#define TILE_F  (128 * PAD)          // padded 128x128 tile (floats)

typedef __attribute__((ext_vector_type(2))) float v2f;
typedef __attribute__((ext_vector_type(8))) float v8f;

__device__ __forceinline__ float sigm(float x) { return 1.0f / (1.0f + __expf(-x)); }

__device__ __forceinline__ v8f wmma_f32(v2f a, v2f b, v8f c) {
  return __builtin_amdgcn_wmma_f32_16x16x4_f32(false, a, false, b,
                                               (short)0, c, false, false);
}

// LayerNorm one 128-float LDS row in place.
__device__ __forceinline__ void rownorm(float* row, const float* __restrict__ w,
                                        const float* __restrict__ b) {
  float s = 0.f;
  #pragma unroll 8
  for (int k = 0; k < 128; ++k) s += row[k];
  float mu = s * 0.0078125f;
  float v = 0.f;
  #pragma unroll 8
  for (int k = 0; k < 128; ++k) { float d = row[k] - mu; v += d * d; }
  float rs = rsqrtf(v * 0.0078125f + 1e-5f);
  #pragma unroll 8
  for (int k = 0; k < 128; ++k) row[k] = (row[k] - mu) * rs * w[k] + b[k];
}

// Copy a row-major 128x128 f32 matrix into PAD-strided LDS (coalesced float4).
__device__ __forceinline__ void stage_pad(const float* __restrict__ W,
                                          float* dst, int tid) {
  for (int i = tid; i < 128 * 32; i += 256) {
    int r = i >> 5, c = (i & 31) << 2;
    float4 v = *(const float4*)(W + (r << 7) + c);
    *(float4*)(dst + r * PAD + c) = v;
  }
}

// Preload a wave-strip's A fragments (row Arow, this lane's k-phase kSub).
__device__ __forceinline__ void load_afrag(const float* __restrict__ Arow,
                                           int kSub, v2f* f) {
  #pragma unroll
  for (int kg = 0; kg < 32; ++kg)
    f[kg] = *(const v2f*)&Arow[(kg << 2) + kSub];
}

// One 16x16 tile: acc = A(preloaded, 16xK) x W[n0:n0+16, :]^T.
// colP = &wt[(n0+l16)*PAD + kSub]; k-group kg at byte offset kg*16.
__device__ __forceinline__ v8f gemm_tile_pre(const v2f* __restrict__ aF,
                                             const float* __restrict__ colP) {
  v8f acc = {};
  #pragma unroll
  for (int kg = 0; kg < 32; ++kg) {
    v2f b = *(const v2f*)&colP[kg << 2];
    acc = wmma_f32(aF[kg], b, acc);
  }
  return acc;
}

// ---------------------------------------------------------------------------
// Kernel A: 64 blocks x 256 thr. 128 active rows per block.
// xn = LN(x[active]); a = sig(xn@WgaT+bga)*(xn@WlaT+bla); b likewise; p = a*b.
// ---------------------------------------------------------------------------
__global__ __launch_bounds__(256) void tri_p_kernel(
    const float* __restrict__ x,
    const float* __restrict__ ln_in_w, const float* __restrict__ ln_in_b,
    const float* __restrict__ Wla, const float* __restrict__ bla,
    const float* __restrict__ Wga, const float* __restrict__ bga,
    const float* __restrict__ Wlb, const float* __restrict__ blb,
    const float* __restrict__ Wgb, const float* __restrict__ bgb,
    float* __restrict__ pbuf) {
  extern __shared__ float sm[];
  float* sm_xn = sm;                 // activations (LN'd)
  float* sm_w0 = sm + TILE_F;        // weight (gate)
  float* sm_w1 = sm + 2 * TILE_F;    // weight (linear)

  const int tid = threadIdx.x;
  const int bb  = blockIdx.x >> 5;           // batch
  const int a0  = (blockIdx.x & 31) << 7;    // first active slot

  // Load 128 active x rows (row = 16*(a0+r)) into LDS, coalesced float4.
  for (int i = tid; i < 128 * 32; i += 256) {
    int r = i >> 5, c = (i & 31) << 2;
    float4 v = *(const float4*)(x + ((size_t)bb * NN + (size_t)(a0 + r) * 16) * DD + c);
    *(float4*)(&sm_xn[r * PAD + c]) = v;
  }
  stage_pad(Wga, sm_w0, tid);
  stage_pad(Wla, sm_w1, tid);
  __syncthreads();
  if (tid < 128) rownorm(&sm_xn[tid * PAD], ln_in_w, ln_in_b);
  __syncthreads();

  const int lane = tid & 31, wv = tid >> 5;
  const int m0 = wv << 4, l16 = lane & 15, lhalf = lane >> 4, kSub = lhalf << 1;

  v2f axF[32];
  load_afrag(&sm_xn[(m0 + l16) * PAD], kSub, axF);
  const int colBase = l16 * PAD + kSub;      // + n0*16*PAD per tile

  v8f a_sav[8];
  #pragma unroll
  for (int n0i = 0; n0i < 8; ++n0i) {
    int cOff = colBase + n0i * 16 * PAD, col = (n0i << 4) + l16;
    v8f g = gemm_tile_pre(axF, sm_w0 + cOff);
    v8f l = gemm_tile_pre(axF, sm_w1 + cOff);
    float bg = bga[col], bl = bla[col];
    v8f av;
    #pragma unroll
    for (int r = 0; r < 8; ++r) av[r] = sigm(g[r] + bg) * (l[r] + bl);
    a_sav[n0i] = av;
  }
  __syncthreads();
  stage_pad(Wgb, sm_w0, tid);
  stage_pad(Wlb, sm_w1, tid);
  __syncthreads();
  #pragma unroll
  for (int n0i = 0; n0i < 8; ++n0i) {
    int cOff = colBase + n0i * 16 * PAD, col = (n0i << 4) + l16;
    v8f g = gemm_tile_pre(axF, sm_w0 + cOff);
    v8f l = gemm_tile_pre(axF, sm_w1 + cOff);
    float bg = bgb[col], bl = blb[col];
    #pragma unroll
    for (int r = 0; r < 8; ++r) {
      float bv   = sigm(g[r] + bg) * (l[r] + bl);
      int   srow = a0 + m0 + r + (lhalf << 3);
      pbuf[((size_t)bb * NACT + srow) * DD + col] = a_sav[n0i][r] * bv;
    }
  }
}

// ---------------------------------------------------------------------------
// Kernel B: 1024 blocks x 256 thr. 128 output rows per block.
// k rows via ring suffix-sum of p; out = sig(xn@WgoT+bgo) * (LN(k)@WloT+blo).
// ---------------------------------------------------------------------------
__global__ __launch_bounds__(256) void tri_out_kernel(
    const float* __restrict__ x,
    const float* __restrict__ ln_in_w, const float* __restrict__ ln_in_b,
    const float* __restrict__ ln_o_w,  const float* __restrict__ ln_o_b,
    const float* __restrict__ Wgo, const float* __restrict__ bgo,
    const float* __restrict__ Wlo, const float* __restrict__ blo,
    const float* __restrict__ pbuf,
    float* __restrict__ out) {
  extern __shared__ float sm[];
  float* sm_xn  = sm;
  float* sm_kn  = sm + TILE_F;
  float* sm_wgo = sm + 2 * TILE_F;
  float* sm_wlo = sm + 3 * TILE_F;

  const int tid  = threadIdx.x;
  const int bb   = blockIdx.x >> 9;           // batch
  const int row0 = (blockIdx.x & 511) << 7;   // first row of this tile

  // Load 128 x rows.
  for (int i = tid; i < 128 * 32; i += 256) {
    int r = i >> 5, c = (i & 31) << 2;
    float4 v = *(const float4*)(x + ((size_t)bb * NN + (size_t)(row0 + r)) * DD + c);
    *(float4*)(&sm_xn[r * PAD + c]) = v;
  }
  stage_pad(Wgo, sm_wgo, tid);
  stage_pad(Wlo, sm_wlo, tid);
  // Build k rows: each of 256 threads does 64 cols of one row.
  {
    int r = tid >> 1, h = (tid & 1) << 6;
    int e = row0 + r;
    int t = e >> 4, d = (e & 15) + 1;
    float4 acc[16];
    #pragma unroll
    for (int j = 0; j < 16; ++j) acc[j] = make_float4(0.f, 0.f, 0.f, 0.f);
    for (int u = d + 1; u <= 16; ++u) {
      const float4* pr = (const float4*)(pbuf +
          ((size_t)bb * NACT + ((t + u) & (NACT - 1))) * DD + h);
      #pragma unroll
      for (int j = 0; j < 16; ++j) {
        float4 v = pr[j];
        acc[j].x += v.x; acc[j].y += v.y; acc[j].z += v.z; acc[j].w += v.w;
      }
    }
    float4* drow = (float4*)(&sm_kn[r * PAD + h]);
    #pragma unroll
    for (int j = 0; j < 16; ++j) drow[j] = acc[j];
  }
  __syncthreads();
  if (tid < 128) {
    rownorm(&sm_xn[tid * PAD], ln_in_w, ln_in_b);
    rownorm(&sm_kn[tid * PAD], ln_o_w, ln_o_b);
  }
  __syncthreads();

  const int lane = tid & 31, wv = tid >> 5;
  const int m0 = wv << 4, l16 = lane & 15, lhalf = lane >> 4, kSub = lhalf << 1;

  v2f axF[32], akF[32];
  load_afrag(&sm_xn[(m0 + l16) * PAD], kSub, axF);
  load_afrag(&sm_kn[(m0 + l16) * PAD], kSub, akF);
  const int colBase = l16 * PAD + kSub;

  #pragma unroll
  for (int n0i = 0; n0i < 8; ++n0i) {
    int cOff = colBase + n0i * 16 * PAD, col = (n0i << 4) + l16;
    const float* bgP = sm_wgo + cOff;
    const float* blP = sm_wlo + cOff;
    v8f accG = {}, accL = {};
    #pragma unroll
    for (int kg = 0; kg < 32; ++kg) {            // interleave the two GEMMs
      v2f bg = *(const v2f*)&bgP[kg << 2];
      accG = wmma_f32(axF[kg], bg, accG);
      v2f bl = *(const v2f*)&blP[kg << 2];
      accL = wmma_f32(akF[kg], bl, accL);
    }
    float bg = bgo[col], bl = blo[col];
    #pragma unroll
    for (int r = 0; r < 8; ++r) {
      int row = row0 + m0 + r + (lhalf << 3);
      out[((size_t)bb * NN + row) * DD + col] = sigm(accG[r] + bg) * (accL[r] + bl);
    }
  }
}

// ---------------------------------------------------------------------------
extern "C" void kernel_launch(void* const* d_in, const int* in_sizes, int n_in,
                              void* d_out, int out_size, void* d_ws, size_t ws_size,
                              hipStream_t stream) {
  const float* x       = (const float*)d_in[0];
  const float* ln_in_w = (const float*)d_in[1];
  const float* ln_in_b = (const float*)d_in[2];
  const float* Wla     = (const float*)d_in[3];
  const float* bla     = (const float*)d_in[4];
  const float* Wga     = (const float*)d_in[5];
  const float* bga     = (const float*)d_in[6];
  const float* Wlb     = (const float*)d_in[7];
  const float* blb     = (const float*)d_in[8];
  const float* Wgb     = (const float*)d_in[9];
  const float* bgb     = (const float*)d_in[10];
  const float* ln_o_w  = (const float*)d_in[11];
  const float* ln_o_b  = (const float*)d_in[12];
  const float* Wgo     = (const float*)d_in[13];
  const float* bgo     = (const float*)d_in[14];
  const float* Wlo     = (const float*)d_in[15];
  const float* blo     = (const float*)d_in[16];
  float* out  = (float*)d_out;
  float* pbuf = (float*)d_ws;                 // 2 * 4096 * 128 f32 = 4 MB

  const int SMEM_A = 3 * TILE_F * (int)sizeof(float);   // ~198 KB
  const int SMEM_B = 4 * TILE_F * (int)sizeof(float);   // ~264 KB (<320 KB/WGP)
  (void)hipFuncSetAttribute((const void*)tri_p_kernel,
      hipFuncAttributeMaxDynamicSharedMemorySize, SMEM_A);
  (void)hipFuncSetAttribute((const void*)tri_out_kernel,
      hipFuncAttributeMaxDynamicSharedMemorySize, SMEM_B);

  tri_p_kernel<<<64, 256, SMEM_A, stream>>>(
      x, ln_in_w, ln_in_b, Wla, bla, Wga, bga, Wlb, blb, Wgb, bgb, pbuf);
  tri_out_kernel<<<1024, 256, SMEM_B, stream>>>(
      x, ln_in_w, ln_in_b, ln_o_w, ln_o_b, Wgo, bgo, Wlo, blo, pbuf, out);
}